// RateModel_30915174597099
// MI455X (gfx1250) — compile-verified
//
#include <hip/hip_runtime.h>
#include <math.h>

// ---------------------------------------------------------------------------
// Rate RNN: h_{t+1} = h_t + ETA*(-h_t + tanh(h_t W_rec^T + b_rec + x_t W_in^T))
// out[t]  = h_{t+1} W_out^T + b_out
// Fully fused persistent-grid scan on 16 WGPs with fp32 WMMA + async LDS DMA.
// ---------------------------------------------------------------------------

typedef __attribute__((ext_vector_type(2))) float v2f;
typedef __attribute__((ext_vector_type(8))) float v8f;

constexpr int   BATCH = 32;
constexpr int   NREC  = 512;
constexpr int   NIN   = 128;
constexpr int   NOUT  = 64;
constexpr float ETA   = 1.0f;
constexpr int   NWG   = 16;   // one 32-column slice of N per workgroup
constexpr int   TPB   = 128;  // 4 waves of 32: one 16x16 WMMA tile per wave

// ---------------------------------------------------------------------------
__global__ void init_state(float* hbuf, unsigned* counter) {
  int i = blockIdx.x * blockDim.x + threadIdx.x;
  if (i < 2 * BATCH * NREC) hbuf[i] = 0.0f;   // s_0 = 0 (and clear pong)
  if (i == 0) *counter = 0u;                  // grid-barrier counter
}

// ---------------------------------------------------------------------------
__launch_bounds__(TPB, 1)
__global__ void rate_scan(const float* __restrict__ x,      // [B][Nt][NIN]
                          const float* __restrict__ W_rec,  // [N][N]
                          const float* __restrict__ b_rec,  // [N]
                          const float* __restrict__ W_in,   // [N][NIN]
                          const float* __restrict__ W_out,  // [NOUT][N]
                          const float* __restrict__ b_out,  // [NOUT]
                          float* __restrict__ out,          // [B][Nt][NOUT]
                          float* __restrict__ hg,           // ping-pong 2*[B][N]
                          unsigned* __restrict__ counter,
                          int NT) {
  // --- LDS (CDNA5: 320KB/WGP; we use ~152KB) -------------------------------
  __shared__ float hl[BATCH * NREC];   // 64KB  full h_t, [b][k]
  __shared__ float Wl2[NREC * 32];     // 64KB  W_rec slice, paired-K layout
  __shared__ float Win2[NIN * 32];     // 16KB  W_in slice, paired-K layout
  __shared__ float woutl[4 * NREC];    //  8KB  W_out slice, [j][k]
  __shared__ float brecl[32];
  __shared__ float boutl[4];

  const int tid  = threadIdx.x;
  const int wg   = blockIdx.x;
  const int n0   = wg * 32;   // this WG's N columns  [n0, n0+32)
  const int o0   = wg * 4;    // this WG's output cols [o0, o0+4)
  const int lane = tid & 31;
  const int wave = tid >> 5;
  const int lo   = lane & 15;   // lane within half-wave
  const int g    = lane >> 4;   // half-wave group (0/1)
  const int mt   = wave & 1;    // M tile (batch rows 16*mt..)
  const int nt   = wave >> 1;   // N tile within the 32-col slice

  // --- One-time weight staging into LDS ------------------------------------
  // Paired-K layout: W2[(k>>1)*64 + c*2 + (k&1)] = W[n0+c][k]
  // -> B fragment (K pair for one column) is a single 8-byte ds_load_b64.
  for (int i = tid; i < NREC * 32; i += TPB) {
    int p = i >> 6, rem = i & 63;
    int c = rem >> 1, par = rem & 1;
    int k = 2 * p + par;
    Wl2[i] = W_rec[(size_t)(n0 + c) * NREC + k];
  }
  for (int i = tid; i < NIN * 32; i += TPB) {
    int p = i >> 6, rem = i & 63;
    int c = rem >> 1, par = rem & 1;
    int k = 2 * p + par;
    Win2[i] = W_in[(size_t)(n0 + c) * NIN + k];
  }
  for (int i = tid; i < 4 * NREC; i += TPB)
    woutl[i] = W_out[(size_t)o0 * NREC + i];
  if (tid < 32) brecl[tid] = b_rec[n0 + tid];
  if (tid < 4)  boutl[tid] = b_out[o0 + tid];
  __syncthreads();

  // Low 32 bits of a generic pointer to LDS == wave-relative LDS byte address
  // (aperture mapping keeps the low 32 bits); used as the async-DMA dest.
  const unsigned hl_lds_base = (unsigned)(uintptr_t)hl;

  unsigned phase = 0;

  // Iterations t=0..NT: load s_t; emit out[t-1]; compute s_{t+1} (t<NT).
  for (int t = 0; t <= NT; ++t) {
    const float* hsrc = hg + (size_t)(t & 1) * BATCH * NREC;
    float*       hdst = hg + (size_t)((t + 1) & 1) * BATCH * NREC;

    // --- stage s_t into LDS via async global->LDS DMA (no VGPR round-trip) --
    {
      // 4096 float4 elements; each thread issues 32 b128 async copies.
      for (int i = tid; i < BATCH * NREC / 4; i += TPB) {
        unsigned     ldsa = hl_lds_base + (unsigned)i * 16u;
        const float* gsrc = hsrc + (size_t)i * 4;
        asm volatile("global_load_async_to_lds_b128 %0, %1, off"
                     :
                     : "v"(ldsa), "v"(gsrc)
                     : "memory");
      }
      // drain this wave's ASYNCcnt; the workgroup barrier below then makes
      // every wave's LDS writes visible WG-wide.
      asm volatile("s_wait_asynccnt 0x0" ::: "memory");
    }
    __syncthreads();

    // --- output projection for time t-1 (hist[t-1] == s_t) -----------------
    if (t >= 1) {
      int b = tid >> 2, j = tid & 3;
      const float4* hr = (const float4*)&hl[b * NREC];
      const float4* wr = (const float4*)&woutl[j * NREC];
      float s = 0.0f;
#pragma unroll 4
      for (int k4 = 0; k4 < NREC / 4; ++k4) {
        float4 hv = hr[k4], wv = wr[k4];
        s += hv.x * wv.x + hv.y * wv.y + hv.z * wv.z + hv.w * wv.w;
      }
      out[((size_t)b * NT + (t - 1)) * NOUT + o0 + j] = s + boutl[j];
    }

    // --- recurrent + input-projection GEMM via fp32 WMMA -------------------
    if (t < NT) {
      v8f acc = {};
      const float* arow = &hl[(mt * 16 + lo) * NREC];  // A row: batch mt*16+lo
      const int    bcol = (nt * 16 + lo) * 2;

      // pre += h_t @ W_rec^T   (K = 512, 128 x v_wmma_f32_16x16x4_f32)
#pragma unroll 8
      for (int k0 = 0; k0 < NREC; k0 += 4) {
        v2f a  = *(const v2f*)&arow[k0 + 2 * g];                 // A[M][k0+2g..+1]
        v2f bf = *(const v2f*)&Wl2[((k0 >> 1) + g) * 64 + bcol]; // B[k0+2g..+1][N]
        acc = __builtin_amdgcn_wmma_f32_16x16x4_f32(false, a, false, bf,
                                                    (short)0, acc, false, false);
      }
      // pre += x_t @ W_in^T    (K = 128, 32 more WMMAs; x stays L2-resident)
      const float* xrow = &x[((size_t)(mt * 16 + lo) * NT + t) * NIN];
#pragma unroll 8
      for (int k0 = 0; k0 < NIN; k0 += 4) {
        v2f a  = *(const v2f*)&xrow[k0 + 2 * g];
        v2f bf = *(const v2f*)&Win2[((k0 >> 1) + g) * 64 + bcol];
        acc = __builtin_amdgcn_wmma_f32_16x16x4_f32(false, a, false, bf,
                                                    (short)0, acc, false, false);
      }

      // --- pointwise update: s_{t+1} = s + ETA*(tanh(pre) - s) -------------
      const int   nloc  = nt * 16 + lo;
      const int   nglob = n0 + nloc;
      const float br    = brecl[nloc];
#pragma unroll
      for (int r = 0; r < 8; ++r) {
        int   bi   = mt * 16 + r + 8 * g;      // C/D layout: VGPR r, half g
        float sold = hl[bi * NREC + nglob];
        float pre  = acc[r] + br;
        float snew = sold + ETA * (tanhf(pre) - sold);
        hdst[bi * NREC + nglob] = snew;
      }
    }

    // --- grid-wide step barrier (monotonic counter, agent-scope fences) ----
    ++phase;
    __syncthreads();
    if (tid == 0) {
      __threadfence();                       // release our h/out stores
      atomicAdd(counter, 1u);
      const unsigned target = (unsigned)NWG * phase;
      while (__hip_atomic_load(counter, __ATOMIC_RELAXED,
                               __HIP_MEMORY_SCOPE_AGENT) < target) {
        __builtin_amdgcn_s_sleep(2);
      }
      __threadfence();                       // acquire: invalidate WGP$
    }
    __syncthreads();
  }
}

// ---------------------------------------------------------------------------
extern "C" void kernel_launch(void* const* d_in, const int* in_sizes, int n_in,
                              void* d_out, int out_size, void* d_ws, size_t ws_size,
                              hipStream_t stream) {
  const float* x     = (const float*)d_in[0];
  const float* W_rec = (const float*)d_in[1];
  const float* b_rec = (const float*)d_in[2];
  const float* W_in  = (const float*)d_in[3];
  const float* W_out = (const float*)d_in[4];
  const float* b_out = (const float*)d_in[5];
  float*       out   = (float*)d_out;

  const int NT = in_sizes[0] / (BATCH * NIN);  // 2000

  // workspace: h ping-pong (128KB) + barrier counter
  float*    hg      = (float*)d_ws;
  unsigned* counter = (unsigned*)((char*)d_ws + (size_t)2 * BATCH * NREC * sizeof(float));

  init_state<<<(2 * BATCH * NREC + 255) / 256, 256, 0, stream>>>(hg, counter);
  rate_scan<<<NWG, TPB, 0, stream>>>(x, W_rec, b_rec, W_in, W_out, b_out,
                                     out, hg, counter, NT);
}